// MovieLensEncoder_64854006170165
// MI455X (gfx1250) — compile-verified
//
#include <hip/hip_runtime.h>

#define N_USERS   100000
#define N_MOVIES  20000
#define N_EDGES   600000
#define FEAT      64
#define HID       128
#define OUT       64
#define N_TUPLES  100000

typedef __attribute__((ext_vector_type(2))) float v2f;
typedef __attribute__((ext_vector_type(8))) float v8f;

// ---------------------------------------------------------------------------
// WMMA helper: acc += A(16xK, row-major, LDS) @ W.T  where W is [ncols][K]
// row-major in global memory (so B[k][n] = W[colbase+n][k]).
// f32 WMMA 16x16x4 fragment layout (ISA 7.12.2):
//   A: lanes0-15 rows M=0..15 K={kb,kb+1}; lanes16-31 same rows K={kb+2,kb+3}
//   B: VGPR0 = rows K=kb (lo lanes) / kb+2 (hi lanes); VGPR1 = kb+1 / kb+3
//   C/D: VGPR j -> row M=j (lo lanes) / M=j+8 (hi lanes), col N = lane&15
// ---------------------------------------------------------------------------
__device__ __forceinline__ v8f wmma_ab(const float* A, int lda,
                                       const float* W, int ldw,
                                       int colbase, int K, v8f acc, int lane) {
  const int rn = lane & 15;          // A row == B col for this lane
  const int kh = (lane >> 4) << 1;   // 0 or 2
  const float* arow = A + rn * lda;
  const float* wrow = W + (colbase + rn) * ldw;
#pragma unroll 4
  for (int kb = 0; kb < K; kb += 4) {
    v2f a, b;
    a.x = arow[kb + kh];
    a.y = arow[kb + kh + 1];
    b.x = wrow[kb + kh];
    b.y = wrow[kb + kh + 1];
    acc = __builtin_amdgcn_wmma_f32_16x16x4_f32(false, a, false, b,
                                                (short)0, acc, false, false);
  }
  return acc;
}

// ---------------------------------------------------------------------------
__global__ __launch_bounds__(256) void zero_int_kernel(int* p, int n) {
  int i = blockIdx.x * 256 + threadIdx.x;
  if (i < n) p[i] = 0;
}

// c1[h] = b1[h] + dot(user_emb, W1r[h])   (x_dst term of conv1 is uniform)
__global__ __launch_bounds__(128) void c1_kernel(const float* emb, const float* W1r,
                                                 const float* b1, float* c1) {
  int h = threadIdx.x;
  float s = b1[h];
  const float* w = W1r + h * HID;
#pragma unroll 8
  for (int k = 0; k < HID; ++k) s += emb[k] * w[k];
  c1[h] = s;
}

// -------------------------- CSR construction ------------------------------
__global__ __launch_bounds__(256) void count_kernel(const int* user_idx,
                                                    const int* movie_idx,
                                                    int* cntu, int* cntm) {
  int e = blockIdx.x * 256 + threadIdx.x;
  if (e >= N_EDGES) return;
  atomicAdd(&cntu[user_idx[e]], 1);
  atomicAdd(&cntm[movie_idx[e]], 1);
}

// block-local exclusive scan + per-block totals
__global__ __launch_bounds__(256) void scan_block_kernel(const int* in, int n,
                                                         int* outx, int* bsums) {
  __shared__ int sh[256];
  int i = blockIdx.x * 256 + threadIdx.x;
  int v = (i < n) ? in[i] : 0;
  sh[threadIdx.x] = v;
  __syncthreads();
  for (int off = 1; off < 256; off <<= 1) {
    int t = (threadIdx.x >= off) ? sh[threadIdx.x - off] : 0;
    __syncthreads();
    sh[threadIdx.x] += t;
    __syncthreads();
  }
  if (i < n) outx[i] = sh[threadIdx.x] - v;          // exclusive
  if (threadIdx.x == 255) bsums[blockIdx.x] = sh[255];
}

// single-block exclusive scan of <=512 block totals
__global__ __launch_bounds__(512) void scan_small_kernel(int* a, int n) {
  __shared__ int sh[512];
  int v = (threadIdx.x < n) ? a[threadIdx.x] : 0;
  sh[threadIdx.x] = v;
  __syncthreads();
  for (int off = 1; off < 512; off <<= 1) {
    int t = (threadIdx.x >= off) ? sh[threadIdx.x - off] : 0;
    __syncthreads();
    sh[threadIdx.x] += t;
    __syncthreads();
  }
  if (threadIdx.x < n) a[threadIdx.x] = sh[threadIdx.x] - v;
}

__global__ __launch_bounds__(256) void scan_add_kernel(int* outx, const int* bsumsx,
                                                       int n) {
  int i = blockIdx.x * 256 + threadIdx.x;
  if (i < n) outx[i] += bsumsx[blockIdx.x];
}

// place each edge's src id into its dst bucket (both directions in one pass)
__global__ __launch_bounds__(256) void fill_kernel(const int* user_idx,
                                                   const int* movie_idx,
                                                   const int* rowu, const int* rowm,
                                                   int* curu, int* curm,
                                                   int* elu, int* elm) {
  int e = blockIdx.x * 256 + threadIdx.x;
  if (e >= N_EDGES) return;
  int u = user_idx[e], m = movie_idx[e];
  elu[rowu[u] + atomicAdd(&curu[u], 1)] = m;
  elm[rowm[m] + atomicAdd(&curm[m], 1)] = u;
}

// -------------------------- gather segment-mean ---------------------------
// float4-vectorized: F4 = F/4 threads per dst row, b128 loads, 2-deep
// neighbor unroll for MLP. block = 256 threads = (256/F4) dst rows.
__global__ __launch_bounds__(256) void gather_mean_kernel(const int* row, const int* cnt,
                                                          const int* elist,
                                                          const float4* src,
                                                          float4* mean,
                                                          int F4, int n_dst) {
  int per = 256 / F4;
  int d = blockIdx.x * per + threadIdx.x / F4;
  int f = threadIdx.x % F4;
  if (d >= n_dst) return;
  int s = row[d];
  int c = cnt[d];
  float4 a0 = make_float4(0.f, 0.f, 0.f, 0.f);
  float4 a1 = make_float4(0.f, 0.f, 0.f, 0.f);
  int i = 0;
  for (; i + 1 < c; i += 2) {
    float4 p = src[(long)elist[s + i] * F4 + f];
    float4 q = src[(long)elist[s + i + 1] * F4 + f];
    a0.x += p.x; a0.y += p.y; a0.z += p.z; a0.w += p.w;
    a1.x += q.x; a1.y += q.y; a1.z += q.z; a1.w += q.w;
  }
  if (i < c) {
    float4 p = src[(long)elist[s + i] * F4 + f];
    a0.x += p.x; a0.y += p.y; a0.z += p.z; a0.w += p.w;
  }
  float cm = fmaxf((float)c, 1.0f);
  float4 r;
  r.x = (a0.x + a1.x) / cm;
  r.y = (a0.y + a1.y) / cm;
  r.z = (a0.z + a1.z) / cm;
  r.w = (a0.w + a1.w) / cm;
  mean[(long)d * F4 + f] = r;
}

// ------------------------------- GEMMs ------------------------------------
// conv1: user_x = relu(mean1 @ W1l.T + c1); 16 user rows/block, 8 waves = 128 cols
__global__ __launch_bounds__(256) void gemm1_kernel(const float* mean1,
                                                    const float* W1l, const float* c1,
                                                    float* user_x) {
  __shared__ alignas(16) float As[16 * FEAT];
  const int rowbase = blockIdx.x * 16;
  // 16*64 floats = 256 float4 -> one b128 per thread
  ((float4*)As)[threadIdx.x] =
      ((const float4*)(mean1 + (long)rowbase * FEAT))[threadIdx.x];
  __syncthreads();
  const int wave = threadIdx.x >> 5, lane = threadIdx.x & 31;
  const int colbase = wave * 16;
  v8f acc = {};
  acc = wmma_ab(As, FEAT, W1l, FEAT, colbase, FEAT, acc, lane);
  const int n = lane & 15, hi = lane >> 4;
  const float addv = c1[colbase + n];
#pragma unroll
  for (int j = 0; j < 8; ++j) {
    float v = fmaxf(acc[j] + addv, 0.0f);
    user_x[(long)(rowbase + j + 8 * hi) * HID + colbase + n] = v;
  }
}

// conv2 fused with lin2: movie_z = relu(mean2@W2l.T + b2 + movie_x@W2r.T);
// then z_movie = movie_z @ Wlin2.T + blin2 -> d_out movie half.
__global__ __launch_bounds__(256) void gemm2_kernel(const float* mean2,
                                                    const float* movie_x,
                                                    const float* W2l, const float* b2,
                                                    const float* W2r,
                                                    const float* Wlin2, const float* blin2,
                                                    float* movie_z, float* Xmovie) {
  __shared__ alignas(16) float Am[16 * HID];
  __shared__ alignas(16) float Ax[16 * FEAT];
  __shared__ alignas(16) float Zt[16 * HID];
  const int rowbase = blockIdx.x * 16;
  {
    const float4* gm = (const float4*)(mean2 + (long)rowbase * HID);
    ((float4*)Am)[threadIdx.x]       = gm[threadIdx.x];          // 512 f4
    ((float4*)Am)[threadIdx.x + 256] = gm[threadIdx.x + 256];
    ((float4*)Ax)[threadIdx.x] =
        ((const float4*)(movie_x + (long)rowbase * FEAT))[threadIdx.x]; // 256 f4
  }
  __syncthreads();
  const int wave = threadIdx.x >> 5, lane = threadIdx.x & 31;
  const int colbase = wave * 16;
  const int n = lane & 15, hi = lane >> 4;
  v8f acc = {};
  acc = wmma_ab(Am, HID, W2l, HID, colbase, HID, acc, lane);
  acc = wmma_ab(Ax, FEAT, W2r, FEAT, colbase, FEAT, acc, lane);
  const float bv = b2[colbase + n];
#pragma unroll
  for (int j = 0; j < 8; ++j) {
    int rr = j + 8 * hi;
    float v = fmaxf(acc[j] + bv, 0.0f);
    movie_z[(long)(rowbase + rr) * HID + colbase + n] = v;
    Zt[rr * HID + colbase + n] = v;
  }
  __syncthreads();
  if (wave < 4) {  // 64 output cols; wave-uniform branch keeps EXEC all-1s
    const int cb2 = wave * 16;
    v8f a2 = {};
    a2 = wmma_ab(Zt, HID, Wlin2, HID, cb2, HID, a2, lane);
    const float bl = blin2[cb2 + n];
#pragma unroll
    for (int j = 0; j < 8; ++j)
      Xmovie[(long)(rowbase + j + 8 * hi) * OUT + cb2 + n] = a2[j] + bl;
  }
}

// conv3 fused with lin1: user_z = relu(mean3@W3l.T + b3 + user_x@W3r.T);
// z_user = user_z @ Wlin1.T + blin1 -> d_out user half.
__global__ __launch_bounds__(256) void gemm3_kernel(const float* mean3,
                                                    const float* user_x,
                                                    const float* W3l, const float* b3,
                                                    const float* W3r,
                                                    const float* Wlin1, const float* blin1,
                                                    float* Xuser) {
  __shared__ alignas(16) float Am[16 * HID];
  __shared__ alignas(16) float Ax[16 * HID];
  __shared__ alignas(16) float Zt[16 * HID];
  const int rowbase = blockIdx.x * 16;
  {
    const float4* gm = (const float4*)(mean3 + (long)rowbase * HID);
    const float4* gx = (const float4*)(user_x + (long)rowbase * HID);
    ((float4*)Am)[threadIdx.x]       = gm[threadIdx.x];
    ((float4*)Am)[threadIdx.x + 256] = gm[threadIdx.x + 256];
    ((float4*)Ax)[threadIdx.x]       = gx[threadIdx.x];
    ((float4*)Ax)[threadIdx.x + 256] = gx[threadIdx.x + 256];
  }
  __syncthreads();
  const int wave = threadIdx.x >> 5, lane = threadIdx.x & 31;
  const int colbase = wave * 16;
  const int n = lane & 15, hi = lane >> 4;
  v8f acc = {};
  acc = wmma_ab(Am, HID, W3l, HID, colbase, HID, acc, lane);
  acc = wmma_ab(Ax, HID, W3r, HID, colbase, HID, acc, lane);
  const float bv = b3[colbase + n];
#pragma unroll
  for (int j = 0; j < 8; ++j)
    Zt[(j + 8 * hi) * HID + colbase + n] = fmaxf(acc[j] + bv, 0.0f);
  __syncthreads();
  if (wave < 4) {
    const int cb2 = wave * 16;
    v8f a2 = {};
    a2 = wmma_ab(Zt, HID, Wlin1, HID, cb2, HID, a2, lane);
    const float bl = blin1[cb2 + n];
#pragma unroll
    for (int j = 0; j < 8; ++j)
      Xuser[(long)(rowbase + j + 8 * hi) * OUT + cb2 + n] = a2[j] + bl;
  }
}

// new_index: row0 = tuples[0], row1 = tuples[1] + N_USERS (values < 2^24, exact in f32)
__global__ __launch_bounds__(256) void index_kernel(const int* tuples, float* out) {
  int t = blockIdx.x * 256 + threadIdx.x;
  if (t >= N_TUPLES) return;
  out[t]            = (float)tuples[t];
  out[N_TUPLES + t] = (float)(tuples[N_TUPLES + t] + N_USERS);
}

// ---------------------------------------------------------------------------
extern "C" void kernel_launch(void* const* d_in, const int* in_sizes, int n_in,
                              void* d_out, int out_size, void* d_ws, size_t ws_size,
                              hipStream_t stream) {
  const float* movie_x  = (const float*)d_in[0];
  const int*   user_idx = (const int*)  d_in[1];
  const int*   movie_idx= (const int*)  d_in[2];
  const int*   tuples   = (const int*)  d_in[3];
  const float* user_emb = (const float*)d_in[4];
  const float* W1l = (const float*)d_in[5];
  const float* b1  = (const float*)d_in[6];
  const float* W1r = (const float*)d_in[7];
  const float* W2l = (const float*)d_in[8];
  const float* b2  = (const float*)d_in[9];
  const float* W2r = (const float*)d_in[10];
  const float* W3l = (const float*)d_in[11];
  const float* b3  = (const float*)d_in[12];
  const float* W3r = (const float*)d_in[13];
  const float* Wlin1 = (const float*)d_in[14];
  const float* blin1 = (const float*)d_in[15];
  const float* Wlin2 = (const float*)d_in[16];
  const float* blin2 = (const float*)d_in[17];

  // ---------------- workspace layout (4-byte units) ----------------
  int* iw = (int*)d_ws;
  int* cntu = iw;                    // 100000   } zeroed
  int* cntm = cntu + N_USERS;        // 20000    } every
  int* curu = cntm + N_MOVIES;       // 100000   } launch
  int* curm = curu + N_USERS;        // 20000    }
  const int zero_ints = 2 * (N_USERS + N_MOVIES);
  int* rowu = curm + N_MOVIES;       // 100000
  int* rowm = rowu + N_USERS;        // 20000
  int* bsum = rowm + N_MOVIES;       // 512 scratch (reused sequentially)
  int* elu  = bsum + 512;            // 600000 (movie srcs bucketed by user dst)
  int* elm  = elu + N_EDGES;         // 600000 (user srcs bucketed by movie dst)

  float* fw     = (float*)(elm + N_EDGES);             // 16B-aligned
  float* mean1  = fw;                                  // 100000*64
  float* mean2  = mean1 + (long)N_USERS * FEAT;        // 20000*128
  float* mean3  = mean2 + (long)N_MOVIES * HID;        // 100000*128
  float* user_x = mean3 + (long)N_USERS * HID;         // 100000*128
  float* movie_z= user_x + (long)N_USERS * HID;        // 20000*128
  float* c1     = movie_z + (long)N_MOVIES * HID;      // 128

  float* X = (float*)d_out;                            // [120000 x 64]
  float* Xuser  = X;
  float* Xmovie = X + (long)N_USERS * OUT;
  float* idx_out = X + (long)(N_USERS + N_MOVIES) * OUT;

  const int egrid  = (N_EDGES + 255) / 256;            // 2344
  const int ublk   = (N_USERS + 255) / 256;            // 391
  const int mblk   = (N_MOVIES + 255) / 256;           // 79

  // 0) zero counts + bucket cursors (ws is not re-poisoned between replays)
  zero_int_kernel<<<(zero_ints + 255) / 256, 256, 0, stream>>>(iw, zero_ints);

  // 1) CSR build: counts -> exclusive scans -> bucket fill
  count_kernel<<<egrid, 256, 0, stream>>>(user_idx, movie_idx, cntu, cntm);
  scan_block_kernel<<<ublk, 256, 0, stream>>>(cntu, N_USERS, rowu, bsum);
  scan_small_kernel<<<1, 512, 0, stream>>>(bsum, ublk);
  scan_add_kernel<<<ublk, 256, 0, stream>>>(rowu, bsum, N_USERS);
  scan_block_kernel<<<mblk, 256, 0, stream>>>(cntm, N_MOVIES, rowm, bsum);
  scan_small_kernel<<<1, 512, 0, stream>>>(bsum, mblk);
  scan_add_kernel<<<mblk, 256, 0, stream>>>(rowm, bsum, N_MOVIES);
  fill_kernel<<<egrid, 256, 0, stream>>>(user_idx, movie_idx, rowu, rowm,
                                         curu, curm, elu, elm);

  // 2) uniform x_dst term of conv1
  c1_kernel<<<1, 128, 0, stream>>>(user_emb, W1r, b1, c1);

  // 3) conv1: gather mean (users <- movies, 64 feats = 16 float4) + WMMA GEMM
  gather_mean_kernel<<<N_USERS / 16, 256, 0, stream>>>(rowu, cntu, elu,
                                                       (const float4*)movie_x,
                                                       (float4*)mean1, FEAT / 4, N_USERS);
  gemm1_kernel<<<N_USERS / 16, 256, 0, stream>>>(mean1, W1l, c1, user_x);

  // 4) conv2: gather mean (movies <- users, 128 feats = 32 float4) + GEMM + lin2
  gather_mean_kernel<<<N_MOVIES / 8, 256, 0, stream>>>(rowm, cntm, elm,
                                                       (const float4*)user_x,
                                                       (float4*)mean2, HID / 4, N_MOVIES);
  gemm2_kernel<<<N_MOVIES / 16, 256, 0, stream>>>(mean2, movie_x, W2l, b2, W2r,
                                                  Wlin2, blin2, movie_z, Xmovie);

  // 5) conv3: gather mean (users <- movies, 128 feats) + GEMM + lin1
  gather_mean_kernel<<<N_USERS / 8, 256, 0, stream>>>(rowu, cntu, elu,
                                                      (const float4*)movie_z,
                                                      (float4*)mean3, HID / 4, N_USERS);
  gemm3_kernel<<<N_USERS / 16, 256, 0, stream>>>(mean3, user_x, W3l, b3, W3r,
                                                 Wlin1, blin1, Xuser);

  // 6) index offset copy
  index_kernel<<<(N_TUPLES + 255) / 256, 256, 0, stream>>>(tuples, idx_out);
}